// ControlFlowScanCDist_22445499089116
// MI455X (gfx1250) — compile-verified
//
#include <hip/hip_runtime.h>
#include <hip/hip_bf16.h>

typedef __attribute__((ext_vector_type(2))) float v2f;
typedef __attribute__((ext_vector_type(8))) float v8f;

#define NROWS 16384
#define DCOLS 32

// Pass 1: sq[row] = ||x[row]||^2. 16384 rows, one thread per row.
__global__ __launch_bounds__(256) void rownorm_kernel(const float* __restrict__ x,
                                                      float* __restrict__ sq) {
  const int row = blockIdx.x * 256 + threadIdx.x;
  const float4* __restrict__ p = (const float4*)(x + (size_t)row * DCOLS);
  float s = 0.0f;
#pragma unroll
  for (int q = 0; q < 8; ++q) {
    float4 v = p[q];
    s = fmaf(v.x, v.x, s);
    s = fmaf(v.y, v.y, s);
    s = fmaf(v.z, v.z, s);
    s = fmaf(v.w, v.w, s);
  }
  sq[row] = s;
}

// Pass 2: each wave computes a 16x64 strip (4 j-tiles). The distance formula
// is folded entirely into the WMMA: A fragments are pre-scaled by -2 and the
// accumulator is initialized with (||a||^2 + ||b||^2), so the 8-step
// V_WMMA_F32_16X16X4_F32 chain yields d2 directly. Epilogue = max + raw
// v_sqrt_f32 + nontemporal store.
__global__ __launch_bounds__(256) void cdist_wmma_kernel(const float* __restrict__ x,
                                                         const float* __restrict__ sq,
                                                         float* __restrict__ out) {
  const int lane = threadIdx.x & 31;
  const int wave = threadIdx.x >> 5;
  const int lo = lane & 15;   // row within A-tile / col within B-tile
  const int hi = lane >> 4;   // half-wave select

  // 8 waves stack vertically: block covers 128 rows x 64 cols.
  const int ti = blockIdx.y * 8 + wave;          // i-tile (0..1023)
  const int col0 = blockIdx.x * 64;              // strip of 4 j-tiles

  // K-permuted ownership: lanes 0-15 own cols 0..15, lanes 16-31 own cols
  // 16..31 of their row -> one contiguous 64B slice per lane (4 x b128 loads).
  const float* __restrict__ aptr = x + (size_t)(ti * 16 + lo) * DCOLS + hi * 16;

  float a[16];
#pragma unroll
  for (int q = 0; q < 4; ++q) {
    float4 va = ((const float4*)aptr)[q];
    a[4 * q + 0] = va.x; a[4 * q + 1] = va.y; a[4 * q + 2] = va.z; a[4 * q + 3] = va.w;
  }
  // Fold the "-2 * dot" factor into A (exact power-of-two scale).
#pragma unroll
  for (int k = 0; k < 16; ++k) a[k] *= -2.0f;

  // Broadcast ||x_row||^2 for the i-tile rows: output row M = r + 8*hi.
  const float sqv = sq[ti * 16 + lo];
  float sqA[8];
#pragma unroll
  for (int r = 0; r < 8; ++r) sqA[r] = __shfl(sqv, r + 8 * hi, 32);

  const float* __restrict__ bbase = x + (size_t)(col0 + lo) * DCOLS + hi * 16;
  float* __restrict__ obase = out + (size_t)(ti * 16 + 8 * hi) * NROWS + (size_t)(col0 + lo);

#pragma unroll
  for (int t = 0; t < 4; ++t) {
    // B-tile fragments (j-tile t), same register layout as A.
    const float* __restrict__ bptr = bbase + (size_t)t * 16 * DCOLS;
    float b[16];
#pragma unroll
    for (int q = 0; q < 4; ++q) {
      float4 vb = ((const float4*)bptr)[q];
      b[4 * q + 0] = vb.x; b[4 * q + 1] = vb.y; b[4 * q + 2] = vb.z; b[4 * q + 3] = vb.w;
    }

    const float sb = sq[col0 + t * 16 + lo];  // ||x_col||^2, col N = lo

    // Accumulator pre-biased with the norm sum: D = (-2A)B^T + (sqA + sqB).
    v8f c;
#pragma unroll
    for (int r = 0; r < 8; ++r) c[r] = sqA[r] + sb;

    // Fragment VGPR v at step s = column (16*hi + 2*s + v); identical
    // K-permutation for A and B (dot product is K-order invariant).
#pragma unroll
    for (int s = 0; s < 8; ++s) {
      v2f av; av.x = a[2 * s]; av.y = a[2 * s + 1];
      v2f bv; bv.x = b[2 * s]; bv.y = b[2 * s + 1];
      c = __builtin_amdgcn_wmma_f32_16x16x4_f32(
          /*neg_a=*/false, av, /*neg_b=*/false, bv,
          /*c_mod=*/(short)0, c, /*reuse_a=*/false, /*reuse_b=*/false);
    }

    // D layout: VGPR r, lane l -> M = r + 8*hi, N = l%16.
    float* __restrict__ ocol = obase + t * 16;
#pragma unroll
    for (int r = 0; r < 8; ++r) {
      float d2 = fmaxf(c[r], 0.0f);
      // Raw v_sqrt_f32: inputs are clamped non-negative normals, ~1ULP.
      __builtin_nontemporal_store(__builtin_amdgcn_sqrtf(d2),
                                  ocol + (size_t)r * NROWS);
    }
  }
}

extern "C" void kernel_launch(void* const* d_in, const int* in_sizes, int n_in,
                              void* d_out, int out_size, void* d_ws, size_t ws_size,
                              hipStream_t stream) {
  const float* x = (const float*)d_in[0];
  float* out = (float*)d_out;
  float* sq = (float*)d_ws;  // 16384 floats = 64 KB scratch

  rownorm_kernel<<<dim3(NROWS / 256), dim3(256), 0, stream>>>(x, sq);

  // Block (8 waves) covers 128 rows x 64 cols; grid covers 16384 x 16384.
  dim3 grid(NROWS / 64, NROWS / 128);
  dim3 block(256);
  cdist_wmma_kernel<<<grid, block, 0, stream>>>(x, sq, out);
}